// Hop2TokenEncoder_9509057593839
// MI455X (gfx1250) — compile-verified
//
#include <hip/hip_runtime.h>

// ---------------------------------------------------------------------------
// Hop2Token encoder for MI455X (gfx1250, wave32).
//
// out[n, 0, :] = x[n, :]
// out[n, h, :] = sum_{edges (n, c)} out_prev[c, :]   for h = 1..3
//
// Pure gather/scatter-add: zero dense FLOPs -> no WMMA (nothing to feed the
// matrix pipes without a per-call sparse re-blocking that would cost more
// traffic than the op itself). The performance levers on this chip are the
// 192 MB L2 (both the source slice and the destination slice of a hop fit),
// L2-executed global_atomic_add_f32, b128 coalesced gathers, scalar loads for
// wave-uniform edge indices, and NT hints to keep the write-once slice 0 from
// polluting L2.
// ---------------------------------------------------------------------------

typedef float v4f __attribute__((ext_vector_type(4)));

#define DFEAT       256
#define NTOK        4                   // 1 + MAX_HOP
#define NODE_STRIDE (DFEAT * NTOK)      // 1024 floats per node in d_out

// ---------------------------------------------------------------------------
// Init: out[:,0,:] = x (non-temporal: write-once, never re-read -> keep out of
// L2), out[:,1..3,:] = 0 (regular stores: these lines are about to be hammered
// by atomics and re-read next hop, we want them resident).
// One thread = one float4 chunk of one node.
// ---------------------------------------------------------------------------
__global__ __launch_bounds__(256)
void hop_init_kernel(const float* __restrict__ x,
                     float* __restrict__ out,
                     int n_nodes) {
    unsigned tid   = blockIdx.x * 256u + threadIdx.x;
    unsigned total = (unsigned)n_nodes * (DFEAT / 4);   // 6.4M float4 chunks
    if (tid >= total) return;

    unsigned n = tid >> 6;     // 64 float4 chunks per node
    unsigned c = tid & 63u;

    const v4f* xp = (const v4f*)x + tid;                // x row stride = 64 v4f
    v4f v = __builtin_nontemporal_load(xp);             // x re-read by hop1 from
                                                        // its own path; NT ok

    v4f* o = (v4f*)out + (size_t)n * (NODE_STRIDE / 4); // 256 v4f per node
    __builtin_nontemporal_store(v, o + c);              // slice 0: NT store
    v4f z = {0.0f, 0.0f, 0.0f, 0.0f};
    o[ 64 + c] = z;                                     // slice 1
    o[128 + c] = z;                                     // slice 2
    o[192 + c] = z;                                     // slice 3
}

// ---------------------------------------------------------------------------
// One hop of scatter-add SpMM: for each edge e: dst[row[e], :] += src[col[e], :]
//
// Mapping: exactly one edge per wave32. Lane L moves float4 chunks L and L+32
// of the 256-float row (2 x global_load_b128, coalesced), then 8 agent-scope
// relaxed f32 atomics (global_atomic_add_f32, executed in L2 where the
// destination slice is resident). Edge id is wave-uniform -> readfirstlane so
// row/col come in via scalar s_load_b64 instead of 32 redundant vector loads.
// ---------------------------------------------------------------------------
__global__ __launch_bounds__(256)
void hop_scatter_kernel(const long long* __restrict__ rows,
                        const long long* __restrict__ cols,
                        const float* __restrict__ src, int src_stride,
                        float* __restrict__ dst,        // pre-offset to slice h
                        int n_edges) {
    unsigned tid = blockIdx.x * 256u + threadIdx.x;
    unsigned e   = tid >> 5;            // one edge per wave32 (whole wave exits
    if (e >= (unsigned)n_edges) return; // together: e is wave-uniform)
    unsigned lane = tid & 31u;

    // Pin the wave-uniform edge id into an SGPR -> scalar index loads.
    unsigned eu  = (unsigned)__builtin_amdgcn_readfirstlane((int)e);
    unsigned row = (unsigned)rows[eu];
    unsigned col = (unsigned)cols[eu];

    const v4f* s = (const v4f*)(src + (size_t)col * (unsigned)src_stride);
    v4f a = s[lane];        // chunks [lane*16B .. ), fully coalesced b128
    v4f b = s[lane + 32];   // second half of the 1KB row

    float* d  = dst + (size_t)row * NODE_STRIDE;
    float* d0 = d + lane * 4u;          // floats [lane*4 .. lane*4+3]
    float* d1 = d0 + 128u;              // floats [lane*4+128 .. ]

    __hip_atomic_fetch_add(&d0[0], a.x, __ATOMIC_RELAXED, __HIP_MEMORY_SCOPE_AGENT);
    __hip_atomic_fetch_add(&d0[1], a.y, __ATOMIC_RELAXED, __HIP_MEMORY_SCOPE_AGENT);
    __hip_atomic_fetch_add(&d0[2], a.z, __ATOMIC_RELAXED, __HIP_MEMORY_SCOPE_AGENT);
    __hip_atomic_fetch_add(&d0[3], a.w, __ATOMIC_RELAXED, __HIP_MEMORY_SCOPE_AGENT);
    __hip_atomic_fetch_add(&d1[0], b.x, __ATOMIC_RELAXED, __HIP_MEMORY_SCOPE_AGENT);
    __hip_atomic_fetch_add(&d1[1], b.y, __ATOMIC_RELAXED, __HIP_MEMORY_SCOPE_AGENT);
    __hip_atomic_fetch_add(&d1[2], b.z, __ATOMIC_RELAXED, __HIP_MEMORY_SCOPE_AGENT);
    __hip_atomic_fetch_add(&d1[3], b.w, __ATOMIC_RELAXED, __HIP_MEMORY_SCOPE_AGENT);
}

// ---------------------------------------------------------------------------
// Launch: init, then 3 serialized hop kernels on the same stream. Hop 1 reads
// the original x (row stride 256); hops 2-3 read the previous out slice
// (row stride 1024). All sizes derived host-side from in_sizes (the num_nodes
// scalar lives in device memory and must not be read during graph capture).
// ---------------------------------------------------------------------------
extern "C" void kernel_launch(void* const* d_in, const int* in_sizes, int n_in,
                              void* d_out, int out_size, void* d_ws, size_t ws_size,
                              hipStream_t stream) {
    const float*     x  = (const float*)d_in[0];
    const long long* ei = (const long long*)d_in[1];   // int64 [2, E]

    const int n_nodes = in_sizes[0] / DFEAT;
    const int n_edges = in_sizes[1] / 2;

    const long long* rows = ei;             // edge_index[0]
    const long long* cols = ei + n_edges;   // edge_index[1]
    float* out = (float*)d_out;             // [N, 4, 256]

    // Init all of d_out (copy slice 0, zero slices 1..3).
    unsigned init_work   = (unsigned)n_nodes * (DFEAT / 4);
    unsigned init_blocks = (init_work + 255u) / 256u;
    hop_init_kernel<<<init_blocks, 256, 0, stream>>>(x, out, n_nodes);

    // 8 edges (8 waves) per 256-thread block.
    unsigned sc_blocks = ((unsigned)n_edges + 7u) / 8u;

    // hop 1: src = x (stride 256) -> slice 1
    hop_scatter_kernel<<<sc_blocks, 256, 0, stream>>>(
        rows, cols, x, DFEAT, out + 1 * DFEAT, n_edges);
    // hop 2: src = slice 1 (stride 1024) -> slice 2
    hop_scatter_kernel<<<sc_blocks, 256, 0, stream>>>(
        rows, cols, out + 1 * DFEAT, NODE_STRIDE, out + 2 * DFEAT, n_edges);
    // hop 3: src = slice 2 (stride 1024) -> slice 3
    hop_scatter_kernel<<<sc_blocks, 256, 0, stream>>>(
        rows, cols, out + 2 * DFEAT, NODE_STRIDE, out + 3 * DFEAT, n_edges);
}